// _GptOssMoELayer_85787676770792
// MI455X (gfx1250) — compile-verified
//
#include <hip/hip_runtime.h>
#include <math.h>

// Problem constants (match reference)
constexpr int   kH     = 2880;
constexpr int   kI     = 2880;
constexpr int   kE     = 8;
constexpr int   kTopK  = 4;
constexpr int   kS     = 64;     // tokens (B=1, S=64)
constexpr float kAlpha = 1.702f;
constexpr float kLimit = 7.0f;

typedef __attribute__((ext_vector_type(2))) float v2f;
typedef __attribute__((ext_vector_type(8))) float v8f;

__device__ __forceinline__ v8f wmma_f32(v2f a, v2f b, v8f c) {
  // 8 args: (neg_a, A, neg_b, B, c_mod, C, reuse_a, reuse_b)
  return __builtin_amdgcn_wmma_f32_16x16x4_f32(false, a, false, b, (short)0, c,
                                               false, false);
}

// ---------------------------------------------------------------------------
// Kernel 1: router — logits = x @ router_w.T + b; top-4; softmax; dense scatter
// One block per token, 256 threads, LDS tree reduction over H.
// ---------------------------------------------------------------------------
__global__ void router_kernel(const float* __restrict__ x,
                              const float* __restrict__ rw,
                              const float* __restrict__ rb,
                              float* __restrict__ dw)   // [kS][kE]
{
  __shared__ float red[kE][256];
  const int t = blockIdx.x;
  const int j = threadIdx.x;
  const float* xr = x + (size_t)t * kH;

  float p[kE];
#pragma unroll
  for (int e = 0; e < kE; ++e) p[e] = 0.0f;

  for (int k = j; k < kH; k += 256) {
    const float xv = xr[k];
#pragma unroll
    for (int e = 0; e < kE; ++e) p[e] += xv * rw[(size_t)e * kH + k];
  }
#pragma unroll
  for (int e = 0; e < kE; ++e) red[e][j] = p[e];
  __syncthreads();

  for (int s = 128; s > 0; s >>= 1) {
    if (j < s) {
#pragma unroll
      for (int e = 0; e < kE; ++e) red[e][j] += red[e][j + s];
    }
    __syncthreads();
  }

  if (j == 0) {
    float logits[kE];
#pragma unroll
    for (int e = 0; e < kE; ++e) logits[e] = red[e][0] + rb[e];

    float vals[kTopK];
    int   idx[kTopK];
    bool  used[kE];
#pragma unroll
    for (int e = 0; e < kE; ++e) used[e] = false;

    for (int r = 0; r < kTopK; ++r) {
      float best = -INFINITY;
      int   bi   = 0;
      for (int e = 0; e < kE; ++e) {
        if (!used[e] && logits[e] > best) { best = logits[e]; bi = e; }
      }
      used[bi] = true;
      vals[r]  = best;
      idx[r]   = bi;
    }
    const float m = vals[0];
    float ex[kTopK], denom = 0.0f;
    for (int r = 0; r < kTopK; ++r) { ex[r] = __expf(vals[r] - m); denom += ex[r]; }

    float o[kE];
#pragma unroll
    for (int e = 0; e < kE; ++e) o[e] = 0.0f;
    for (int r = 0; r < kTopK; ++r) o[idx[r]] = ex[r] / denom;
#pragma unroll
    for (int e = 0; e < kE; ++e) dw[t * kE + e] = o[e];
  }
}

// ---------------------------------------------------------------------------
// Kernel 2: gate/up GEMMs + fused clipped-SwiGLU epilogue, router weight folded
// into the activation so the down projection becomes a plain expert-sum.
// grid = (kI/16, kE), block = 32 (ONE wave). The wave owns ALL 4 M-tiles
// (64 tokens) and one 16-wide N-tile for BOTH gate and up: every weight
// fragment is loaded exactly once (no cross-wave duplication); A fragments
// come from the L2-resident x via b64 loads.
// Fragment layouts (V_WMMA_F32_16X16X4_F32):
//   A (16x4): lanes 0-15 hold K0/K1 for row M=lane; lanes 16-31 hold K2/K3.
//   B (4x16): lanes 0-15 hold K0/K1 for col N=lane; lanes 16-31 hold K2/K3.
//   C (16x16): lane N = lane&15; VGPR r -> row M = r (+8 for lanes 16-31).
// ---------------------------------------------------------------------------
__global__ void gateup_kernel(const float* __restrict__ x,   // [kS][kH]
                              const float* __restrict__ Wg,  // [kE][kH][kI]
                              const float* __restrict__ bg,  // [kE][kI]
                              const float* __restrict__ Wu,
                              const float* __restrict__ bu,
                              const float* __restrict__ dw,  // [kS][kE]
                              float* __restrict__ act)       // [kE][kS][kI]
{
  const int e    = blockIdx.y;
  const int n0   = blockIdx.x * 16;
  const int lane = threadIdx.x & 31;
  const int lr   = lane & 15;
  const int hi   = lane >> 4;          // K half: 0 -> K0/K1, 1 -> K2/K3

  const float* xr = x + (size_t)lr * kH;     // rows lr, lr+16, lr+32, lr+48
  const size_t wb = (size_t)e * kH * kI;
  const float* g0 = Wg + wb + n0 + lr;
  const float* u0 = Wu + wb + n0 + lr;

  v8f cg[4] = {{}, {}, {}, {}};
  v8f cu[4] = {{}, {}, {}, {}};

#pragma unroll 2
  for (int k = 0; k < kH; k += 4) {
    const int    ka = k + hi * 2;
    const size_t r0 = (size_t)ka * kI;
    const size_t r1 = (size_t)(ka + 1) * kI;
    v2f bG; bG.x = g0[r0]; bG.y = g0[r1];
    v2f bU; bU.x = u0[r0]; bU.y = u0[r1];
    // Prefetch the weight stream 32 rows ahead (global_prefetch_b8: no
    // LOADcnt, no VGPR cost -> extra HBM misses in flight for free).
    if (k + 34 < kH) {
      __builtin_prefetch(g0 + (size_t)(ka + 32) * kI, 0, 1);
      __builtin_prefetch(u0 + (size_t)(ka + 32) * kI, 0, 1);
    }
#pragma unroll
    for (int m = 0; m < 4; ++m) {
      v2f a = *(const v2f*)(xr + (size_t)m * 16 * kH + ka);  // b64, 8B aligned
      cg[m] = wmma_f32(a, bG, cg[m]);
      cu[m] = wmma_f32(a, bU, cu[m]);
    }
  }

  // Epilogue: bias, clamp, SwiGLU, scale by router weight dw[t][e].
  const int   i   = n0 + lr;
  const float bgv = bg[e * kI + i];
  const float buv = bu[e * kI + i];
  float* abase = act + (size_t)e * kS * kI;

#pragma unroll
  for (int m = 0; m < 4; ++m) {
#pragma unroll
    for (int r = 0; r < 8; ++r) {
      const int   t = m * 16 + r + hi * 8;
      const float w = dw[t * kE + e];
      float g = fminf(cg[m][r] + bgv, kLimit);
      float u = fminf(fmaxf(cu[m][r] + buv, -kLimit), kLimit);
      float glu = g / (1.0f + __expf(-kAlpha * g));
      abase[(size_t)t * kI + i] = w * (u + 1.0f) * glu;
    }
  }
}

// ---------------------------------------------------------------------------
// Kernel 3: per-expert down projection into partial buffers.
// grid = (kH/16, kE), block = 32 (one wave, 4 M-tiles x one 16-wide H-tile).
// act is already scaled by dw, so partials just need summing later.
// 1440 independent weight streams maximize HBM MLP; prefetch 32 rows ahead.
// ---------------------------------------------------------------------------
__global__ void down_kernel(const float* __restrict__ act,   // [kE][kS][kI]
                            const float* __restrict__ Wd,    // [kE][kI][kH]
                            float* __restrict__ part)        // [kE][kS][kH]
{
  const int e    = blockIdx.y;
  const int h0   = blockIdx.x * 16;
  const int lane = threadIdx.x & 31;
  const int lr   = lane & 15;
  const int hi   = lane >> 4;

  const float* ar = act + ((size_t)e * kS + lr) * kI;
  const float* d0 = Wd + (size_t)e * kI * kH + h0 + lr;

  v8f c[4] = {{}, {}, {}, {}};

#pragma unroll 2
  for (int k = 0; k < kI; k += 4) {
    const int    ka = k + hi * 2;
    const size_t r0 = (size_t)ka * kH;
    const size_t r1 = (size_t)(ka + 1) * kH;
    v2f b; b.x = d0[r0]; b.y = d0[r1];
    if (k + 34 < kI) {
      __builtin_prefetch(d0 + (size_t)(ka + 32) * kH, 0, 1);
    }
#pragma unroll
    for (int m = 0; m < 4; ++m) {
      v2f a = *(const v2f*)(ar + (size_t)m * 16 * kI + ka);
      c[m] = wmma_f32(a, b, c[m]);
    }
  }

  const int h = h0 + lr;
  float* pb = part + (size_t)e * kS * kH;
#pragma unroll
  for (int m = 0; m < 4; ++m) {
#pragma unroll
    for (int r = 0; r < 8; ++r) {
      const int t = m * 16 + r + hi * 8;
      pb[(size_t)t * kH + h] = c[m][r];
    }
  }
}

// ---------------------------------------------------------------------------
// Kernel 4: deterministic combine — out = sum_e part[e] + sum_e dw[t,e]*bd[e].
// ---------------------------------------------------------------------------
__global__ void combine_kernel(const float* __restrict__ part, // [kE][kS][kH]
                               const float* __restrict__ bd,   // [kE][kH]
                               const float* __restrict__ dw,   // [kS][kE]
                               float* __restrict__ out)        // [kS][kH]
{
  const int idx = blockIdx.x * 256 + threadIdx.x;  // t*kH + h
  if (idx >= kS * kH) return;
  const int t = idx / kH;
  const int h = idx - t * kH;
  float s = 0.0f;
#pragma unroll
  for (int e = 0; e < kE; ++e) {
    s += part[(size_t)e * kS * kH + idx] + dw[t * kE + e] * bd[e * kH + h];
  }
  out[idx] = s;
}

// ---------------------------------------------------------------------------
extern "C" void kernel_launch(void* const* d_in, const int* in_sizes, int n_in,
                              void* d_out, int out_size, void* d_ws, size_t ws_size,
                              hipStream_t stream) {
  const float* x        = (const float*)d_in[0]; // hidden_states [1,64,H]
  const float* router_w = (const float*)d_in[1]; // [E,H]
  const float* router_b = (const float*)d_in[2]; // [E]
  const float* Wg       = (const float*)d_in[3]; // [E,H,I]
  const float* bg       = (const float*)d_in[4]; // [E,I]
  const float* Wu       = (const float*)d_in[5]; // [E,H,I]
  const float* bu       = (const float*)d_in[6]; // [E,I]
  const float* Wd       = (const float*)d_in[7]; // [E,I,H]
  const float* bd       = (const float*)d_in[8]; // [E,H]
  float*       out      = (float*)d_out;         // [1,64,H]

  // Workspace layout.
  float* dw   = (float*)d_ws;                                   // 2 KB
  float* act  = (float*)((char*)d_ws + 4096);                   // kE*kS*kI (5.9 MB)
  float* part = (float*)((char*)d_ws + 4096 +
                         (size_t)kE * kS * kI * sizeof(float)); // kE*kS*kH (5.9 MB)

  router_kernel<<<dim3(kS), dim3(256), 0, stream>>>(x, router_w, router_b, dw);
  gateup_kernel<<<dim3(kI / 16, kE), dim3(32), 0, stream>>>(x, Wg, bg, Wu, bu,
                                                            dw, act);
  down_kernel<<<dim3(kH / 16, kE), dim3(32), 0, stream>>>(act, Wd, part);
  combine_kernel<<<dim3((kS * kH + 255) / 256), dim3(256), 0, stream>>>(part, bd,
                                                                        dw, out);
}